// FixedPointLayer_41308995452955
// MI455X (gfx1250) — compile-verified
//
#include <hip/hip_runtime.h>
#include <hip/hip_bf16.h>

// Problem constants (match reference)
#define HEADS 8
#define NDIM  2048
#define GAMMA 0.9f
#define TOLSQ (1e-6f * 1e-6f)
#define MAX_ITER 50

typedef __attribute__((ext_vector_type(2))) float v2f;
typedef __attribute__((ext_vector_type(8))) float v8f;

// Block: 512 threads = 16 waves.
//   wave w: row-group g = w&3 (16 rows each -> 64 rows/block),
//           k-part    p = w>>2 (512 K each -> K=2048)
// Grid: 8 heads * 32 row-blocks = 256 blocks.
//
// WMMA usage: V_WMMA_F32_16X16X4_F32, A fragment loaded straight from global
// in the ISA-documented 16x4 f32 layout (lane L: row L&15, k-pair (L>>4)*2),
// B fragment = x[k..k+3] broadcast across all 16 N-columns using the SAME
// per-lane float2 pattern, so every C column holds the same (correct) dot
// product -> results live in lanes 0 (rows 0..7) and 16 (rows 8..15).
// Per-step b64 loads pipeline deeply (compiler keeps ~16-18 in flight), which
// is what actually hides L2 latency on this bandwidth-bound stream.

template <bool FINAL>
__global__ __launch_bounds__(512) void fp_iter(
    const float* __restrict__ A, const float* __restrict__ bvec,
    const float* __restrict__ x_in, float* __restrict__ x_out,
    float* __restrict__ partials, const int* __restrict__ flags, int t)
{
    const int tid     = threadIdx.x;
    const int head    = blockIdx.x >> 5;
    const int rowBase = (blockIdx.x & 31) * 64;

    if (!FINAL) {
        // Converged earlier: pass x through unchanged (reference keeps old x).
        const bool conv = (t > 0) && (flags[t - 1] != 0);
        if (conv) {
            if (tid < 64) {
                const int idx = head * NDIM + rowBase + tid;
                x_out[idx] = x_in[idx];
            }
            if (tid == 0) partials[blockIdx.x] = 0.0f;
            return;
        }
    }

    const int wave = tid >> 5;
    const int lane = tid & 31;
    const int g    = wave & 3;          // row group within block
    const int p    = wave >> 2;         // K partition (0..3)
    const int m    = lane & 15;         // row within 16-row tile
    const int kk   = (lane >> 4) * 2;   // k-pair select (A & B layouts)

    const float* aPtr = A + (size_t)head * NDIM * NDIM
                          + (size_t)(rowBase + g * 16 + m) * NDIM
                          + p * 512 + kk;
    const float* xPtr = x_in + head * NDIM + p * 512 + kk;

    v8f c0 = {};
    v8f c1 = {};
    // 128 K-steps of 4 (covers this wave's 512-wide K partition).
    for (int blk = 0; blk < 16; ++blk) {
        // WGP-scope prefetch (default scope = 0): pulls the L2-resident A
        // stream into L0/WGP$ ahead of the scattered WMMA-layout reads.
        // (__builtin_prefetch can only reach SE/SYS scope = GL2-only.)
        asm volatile("global_prefetch_b8 %0, off offset:512"
                     :: "v"(aPtr) : "memory");
        #pragma unroll
        for (int u = 0; u < 4; ++u) {
            v2f a0 = *(const v2f*)(aPtr);
            v2f b0 = *(const v2f*)(xPtr);
            v2f a1 = *(const v2f*)(aPtr + 4);
            v2f b1 = *(const v2f*)(xPtr + 4);
            c0 = __builtin_amdgcn_wmma_f32_16x16x4_f32(
                false, a0, false, b0, (short)0, c0, false, false);
            c1 = __builtin_amdgcn_wmma_f32_16x16x4_f32(
                false, a1, false, b1, (short)0, c1, false, false);
            aPtr += 8;
            xPtr += 8;
        }
    }
    const v8f c = c0 + c1;

    // Combine the 4 K-partitions through LDS.
    __shared__ float part[4][64];
    if (lane == 0) {
        #pragma unroll
        for (int r = 0; r < 8; ++r) part[p][g * 16 + r] = c[r];
    } else if (lane == 16) {
        #pragma unroll
        for (int r = 0; r < 8; ++r) part[p][g * 16 + 8 + r] = c[r];
    }
    __syncthreads();

    __shared__ float dsq[64];
    if (tid < 64) {
        const int row = rowBase + tid;
        const int idx = head * NDIM + row;
        const float sum = part[0][tid] + part[1][tid] + part[2][tid] + part[3][tid];
        const float y = tanhf(GAMMA * sum + bvec[idx]);
        if (FINAL) {
            x_out[idx] = y;                       // x_out == d_out here
        } else {
            const float d = x_in[idx] - y;
            x_out[idx] = y;
            dsq[tid] = d * d;
        }
    }
    if (!FINAL) {
        __syncthreads();
        if (tid == 0) {
            float s = 0.0f;
            #pragma unroll
            for (int i = 0; i < 64; ++i) s += dsq[i];   // deterministic order
            partials[blockIdx.x] = s;
        }
    }
}

// Single-block deterministic reduction of the 256 block partials; sets the
// convergence flag and, on the triggering iteration, restores pre-update x
// (reference breaks BEFORE accepting x_next).
__global__ __launch_bounds__(256) void fp_check(
    const float* __restrict__ partials, int t, int* __restrict__ flags,
    const float* __restrict__ x_in, float* __restrict__ x_out)
{
    __shared__ float s[256];
    __shared__ int convNow;
    const int tid = threadIdx.x;
    s[tid] = partials[tid];
    __syncthreads();
    if (tid == 0) {
        float sum = 0.0f;
        for (int i = 0; i < 256; ++i) sum += s[i];
        const int prev = (t > 0) ? flags[t - 1] : 0;
        const int now  = (!prev) && (sum < TOLSQ);
        flags[t] = prev | now;
        convNow  = now;
    }
    __syncthreads();
    if (convNow) {
        for (int i = tid; i < HEADS * NDIM; i += 256) x_out[i] = x_in[i];
    }
}

extern "C" void kernel_launch(void* const* d_in, const int* in_sizes, int n_in,
                              void* d_out, int out_size, void* d_ws, size_t ws_size,
                              hipStream_t stream)
{
    (void)in_sizes; (void)n_in; (void)out_size; (void)ws_size;
    const float* A  = (const float*)d_in[0];   // [8,2048,2048] f32
    const float* bv = (const float*)d_in[1];   // [8,2048,1]    f32

    // Workspace layout (needs 132,352 bytes):
    //   [0, 64K)            xbuf0    (must start zeroed: x0 = 0)
    //   [64K, 64K+256)      flags    (50 ints, must start zeroed)
    //   [64K+256, 128K+256) xbuf1    (fully written by iter 0)
    //   [128K+256, +1K)     partials (256 f32, overwritten each iter)
    char* ws = (char*)d_ws;
    float* xbuf0    = (float*)ws;
    int*   flags    = (int*)(ws + 65536);
    float* xbuf1    = (float*)(ws + 65536 + 256);
    float* partials = (float*)(ws + 65536 + 256 + 65536);

    hipMemsetAsync(d_ws, 0, 65536 + 256, stream);   // x0 = 0, flags = 0

    float* xs[2] = { xbuf0, xbuf1 };
    for (int t = 0; t < MAX_ITER; ++t) {
        fp_iter<false><<<256, 512, 0, stream>>>(
            A, bv, xs[t & 1], xs[(t + 1) & 1], partials, flags, t);
        fp_check<<<1, 256, 0, stream>>>(
            partials, t, flags, xs[t & 1], xs[(t + 1) & 1]);
    }
    // One final differentiable step: z = tanh(gamma*A@z* + b) -> d_out [8,2048]
    fp_iter<true><<<256, 512, 0, stream>>>(
        A, bv, xbuf0 /* z* after 50 steps (even parity) */, (float*)d_out,
        nullptr, nullptr, 0);
}